// DensityPooling_22213570855423
// MI455X (gfx1250) — compile-verified
//
#include <hip/hip_runtime.h>
#include <hip/hip_bf16.h>

// DensityPooling for MI455X (gfx1250, wave32).
// Phase 1: fused exp+weighted-pool (trans-bound: 64M v_exp_f32), deterministic
//          per-slice partials in d_ws (needs 4 MB scratch).
// Phase 2: phi = ln(pool+eps), then (512x32)x(32x256) lift via
//          V_WMMA_F32_16X16X4_F32 (full fp32 on the matrix pipe, K chained 8x).

#define NB     32        // n_basis (= GEMM K)
#define BATCH  8
#define XDIM   4096
#define ADIM   64
#define EDIM   256
#define CX     256                 // x-values per chunk
#define NCHUNK (XDIM / CX)         // 16
#define XG     4                   // x sub-groups per block
#define NSLICE (NCHUNK * XG)       // 64 partial slices
#define ROWS   (BATCH * ADIM)      // 512 GEMM rows (b*64+a)

typedef __attribute__((ext_vector_type(2))) float v2f;
typedef __attribute__((ext_vector_type(8))) float v8f;

// Hardware transcendentals: v_exp_f32 is exp2, v_log_f32 is log2.
__device__ __forceinline__ float hw_exp2(float x) { return __builtin_amdgcn_exp2f(x); }
__device__ __forceinline__ float hw_log2(float x) { return __builtin_amdgcn_logf(x); }

// ---------------------------------------------------------------------------
// Kernel 1: pooled partials.
//   partial[slice][row][s] = norms[s] * sum_{x in slice} wrho[b,x]*exp(-g[s]*d^2)
// lane -> a (coalesced loads of distances), xg -> x sub-stream.
// ---------------------------------------------------------------------------
__global__ void dp_pool_kernel(const float* __restrict__ wrho,
                               const float* __restrict__ dist,
                               const float* __restrict__ gammas,
                               float* __restrict__ partial) {
  const int b     = blockIdx.x;        // 0..7
  const int chunk = blockIdx.y;        // 0..15
  const int tid   = threadIdx.x;       // 0..255
  const int a     = tid & (ADIM - 1);  // 0..63
  const int xg    = tid >> 6;          // 0..3
  const int x0    = chunk * CX;

  // -gammas[s] * log2(e): v_exp_f32 computes exp2, so fold the base change in.
  float mg[NB];
#pragma unroll
  for (int s = 0; s < NB; ++s)
    mg[s] = -gammas[s] * 1.4426950408889634f;

  float acc[NB];
#pragma unroll
  for (int s = 0; s < NB; ++s) acc[s] = 0.0f;

  const float* drow = dist + (size_t)b * XDIM * ADIM + a;
  const float* wrow = wrho + (size_t)b * XDIM;

  for (int x = x0 + xg; x < x0 + CX; x += XG) {
    float d  = drow[(size_t)x * ADIM];   // 128B-coalesced across a-lanes
    float wr = wrow[x];                  // wave-uniform, cache hit
    float d2 = d * d;
#pragma unroll
    for (int s = 0; s < NB; ++s)
      acc[s] = fmaf(wr, hw_exp2(mg[s] * d2), acc[s]);  // 1 mul + 1 trans + 1 fmac
  }

  const int slice = chunk * XG + xg;               // 0..63
  const int row   = b * ADIM + a;                  // 0..511
  float* outp = partial + ((size_t)slice * ROWS + row) * NB;
#pragma unroll
  for (int s = 0; s < NB; ++s) {
    // norms[s] = (pi/gamma)^1.5, applied once per output (not in hot loop)
    float g  = gammas[s];
    float nr = hw_exp2(1.5f * hw_log2(3.14159265358979323846f / g));
    outp[s] = nr * acc[s];
  }
}

// ---------------------------------------------------------------------------
// Kernel 2: phi = ln(sum_slices partial + eps); out = phi @ W^T via
// V_WMMA_F32_16X16X4_F32.  Block: 512 thr (16 waves), covers 32 rows x 128
// cols => 16 tiles of 16x16, one tile per wave, K=32 chained as 8 wmma steps.
// ---------------------------------------------------------------------------
__global__ void dp_lift_kernel(const float* __restrict__ partial,
                               const float* __restrict__ W,
                               float* __restrict__ out) {
  const int rg  = blockIdx.x;   // 0..15 : 32-row group
  const int cg  = blockIdx.y;   // 0..1  : 128-col group
  const int tid = threadIdx.x;  // 0..511

  __shared__ float phiS[32 * 33];   // [row_local][s], padded
  __shared__ float WS[NB * 132];    // [s][col_local], padded: WS[s][c]=W[c0+c][s]

  // Build phi tile: deterministic fixed-order reduction over the 64 slices.
  for (int e = tid; e < 32 * NB; e += 512) {
    int r = e >> 5, s = e & 31;
    int rowg = rg * 32 + r;
    float sum = 0.0f;
    for (int sl = 0; sl < NSLICE; ++sl)
      sum += partial[((size_t)sl * ROWS + rowg) * NB + s];
    phiS[r * 33 + s] = hw_log2(sum + 1e-4f) * 0.6931471805599453f; // ln
  }
  // Transposed W tile.
  for (int e = tid; e < NB * 128; e += 512) {
    int s = e >> 7, c = e & 127;
    WS[s * 132 + c] = W[((size_t)(cg * 128 + c)) * NB + s];
  }
  __syncthreads();

  const int wave = tid >> 5;       // wave32
  const int lane = tid & 31;
  const int half = lane >> 4;      // 0: lanes 0-15, 1: lanes 16-31
  const int l16  = lane & 15;
  const int mt   = wave & 1;       // row tile in block (0..1)
  const int nt   = wave >> 1;      // col tile in block (0..7)
  const int arow = mt * 16 + l16;  // local phi row for A fragment
  const int bcol = nt * 16 + l16;  // local W col for B fragment

  v8f c = {};
#pragma unroll
  for (int kk = 0; kk < NB; kk += 4) {
    // A 16x4 (ISA 7.12.2): lanes 0-15: v0=K+0, v1=K+1 ; lanes 16-31: v0=K+2, v1=K+3
    v2f av, bv;
    av.x = phiS[arow * 33 + kk + 2 * half];
    av.y = phiS[arow * 33 + kk + 2 * half + 1];
    // B 4x16: rows striped across lanes within a VGPR (layout assumed symmetric:
    // v0 = rows K+0/K+1 on lane halves, v1 = rows K+2/K+3). Not HW-verifiable here.
    bv.x = WS[(kk + half) * 132 + bcol];
    bv.y = WS[(kk + 2 + half) * 132 + bcol];
    c = __builtin_amdgcn_wmma_f32_16x16x4_f32(
        /*neg_a=*/false, av, /*neg_b=*/false, bv,
        /*c_mod=*/(short)0, c, /*reuse_a=*/false, /*reuse_b=*/false);
  }

  // C/D layout: VGPR r -> row r (lanes 0-15) / row r+8 (lanes 16-31), N = l16.
  const int row0 = rg * 32 + mt * 16 + half * 8;
  const int col  = cg * 128 + nt * 16 + l16;
#pragma unroll
  for (int r = 0; r < 8; ++r)
    out[(size_t)(row0 + r) * EDIM + col] = c[r];
}

// ---------------------------------------------------------------------------
extern "C" void kernel_launch(void* const* d_in, const int* in_sizes, int n_in,
                              void* d_out, int out_size, void* d_ws, size_t ws_size,
                              hipStream_t stream) {
  const float* wrho   = (const float*)d_in[0];  // (8, 4096)
  const float* dist   = (const float*)d_in[1];  // (8, 4096, 64)
  const float* gammas = (const float*)d_in[2];  // (32,)
  const float* W      = (const float*)d_in[3];  // (256, 32)
  float* out     = (float*)d_out;               // (8, 64, 256)
  float* partial = (float*)d_ws;                // NSLICE*ROWS*NB*4 = 4 MB

  dim3 g1(BATCH, NCHUNK);          // 128 blocks x 8 waves
  dp_pool_kernel<<<g1, 256, 0, stream>>>(wrho, dist, gammas, partial);

  dim3 g2(ROWS / 32, EDIM / 128);  // 16 x 2 blocks x 16 waves
  dp_lift_kernel<<<g2, 512, 0, stream>>>(partial, W, out);
}